// Map3D_42554535968833
// MI455X (gfx1250) — compile-verified
//
#include <hip/hip_runtime.h>

// ---------------- problem constants (match reference) ----------------
#define SZ      64
#define DS      60
#define PW      68                  // padded width for dims 1,2 (SZ + 4)
#define SVOL    (64 * PW * PW)      // padded scalar-field volume = 295936
#define FC_IN   (DS * SZ * SZ)      // 245760
#define BATCH   4
#define MPAD    16                  // batch padded to WMMA M
#define KSPLIT  64
#define KCHUNK  (FC_IN / KSPLIT)    // 3840
#define CONVEY  0.9f

typedef __attribute__((ext_vector_type(2))) float v2f;
typedef __attribute__((ext_vector_type(4))) float v4f;
typedef __attribute__((ext_vector_type(8))) float v8f;

#define WMMA_F32(a, b, c) \
    __builtin_amdgcn_wmma_f32_16x16x4_f32(false, (a), false, (b), (short)0, (c), false, false)

// ---------------- phase 1: recurrent scalar field ----------------
// s[p][q][r] = sum over 4 channels of padded vinput  (p<64, q,r<68)
__global__ void init_s_kernel(const float* __restrict__ vin, float* __restrict__ s)
{
    int idx = blockIdx.x * blockDim.x + threadIdx.x;
    if (idx >= SVOL) return;
    int r = idx % PW, t = idx / PW;
    int q = t % PW,   p = t / PW;
    float v = 0.f;
    if (p >= 2 && p < DS + 2 && q >= 2 && q < SZ + 2 && r >= 2 && r < SZ + 2) {
        const float* c4 = vin + (size_t)(((p - 2) * SZ + (q - 2)) * SZ + (r - 2)) * 4;
        v = c4[0] + c4[1] + c4[2] + c4[3];
    }
    s[idx] = v;
}

// one recurrence step: s_out = 4 * relu(core) * CONVEY inside [0:60,0:64,0:64], else 0
__global__ void step_kernel(const float* __restrict__ s_in,
                            const float* __restrict__ syn,
                            float* __restrict__ s_out)
{
    int idx = blockIdx.x * blockDim.x + threadIdx.x;
    if (idx >= SVOL) return;
    int r = idx % PW, t = idx / PW;
    int q = t % PW,   p = t / PW;
    float val = 0.f;
    if (p < DS && q < SZ && r < SZ) {
        const float* sy = syn + (size_t)((p * SZ + q) * SZ + r) * 64;
        float core = 0.f;
#pragma unroll
        for (int a = 0; a < 4; ++a)
#pragma unroll
            for (int b = 0; b < 4; ++b) {
                const float* srow = s_in + ((p + a) * PW + (q + b)) * PW + r;
#pragma unroll
                for (int c = 0; c < 4; ++c)
                    core = fmaf(sy[(a * 4 + b) * 4 + c], srow[c], core);
            }
        val = fmaxf(core, 0.f) * (CONVEY * 4.0f);   // store ns = 4*v
    }
    s_out[idx] = val;
}

// materialize zero-padded (16 x FC_IN) activation matrix for FC1
__global__ void make_x_kernel(const float* __restrict__ s_fin, float* __restrict__ xb)
{
    int g = blockIdx.x * blockDim.x + threadIdx.x;
    if (g >= MPAD * FC_IN) return;
    int row = g / FC_IN;
    float v = 0.f;
    if (row < BATCH) {
        int gf  = g;                       // flat index into (60,64,64,4)
        int i   = gf / (SZ * SZ * 4);
        int rem = gf % (SZ * SZ * 4);
        int j   = rem / (SZ * 4);
        int kk  = (rem % (SZ * 4)) / 4;
        if (i + 2 < DS && j + 2 < SZ && kk + 2 < SZ)
            v = 0.25f * s_fin[((i + 2) * PW + (j + 2)) * PW + (kk + 2)];
    }
    xb[g] = v;
}

// ---------------- phase 2: MLP ----------------
__global__ void fc1_init_kernel(const float* __restrict__ b1, float* __restrict__ y1)
{
    int idx = blockIdx.x * blockDim.x + threadIdx.x;
    if (idx < BATCH * 1024) y1[idx] = b1[idx & 1023];
}

// FC1: y1 += x(16 x K) * W1^T tile, f32 WMMA 16x16x4, K split across waves.
// grid (8, 64), block 256 (8 waves). wave -> ksplit, blockIdx.y -> n-tile.
//
// K-order trick: the contraction is order-invariant, so each lane loads a
// float4 at (k0 + 4*half) and we issue two WMMAs per load pair:
//   #1 logical K0..K3 = global {k0, k0+1, k0+4, k0+5}  -> .xy of each lane
//   #2 logical K0..K3 = global {k0+2,k0+3,k0+6, k0+7}  -> .zw of each lane
// This matches the ISA A/B layout (VGPR0 = K0|K2 by half-wave, VGPR1 = K1|K3).
__global__ void fc1_wmma_kernel(const float* __restrict__ xb,
                                const float* __restrict__ W1,
                                float* __restrict__ y1)
{
    const int lane   = threadIdx.x & 31;
    const int wave   = threadIdx.x >> 5;
    const int ksplit = blockIdx.x * 8 + wave;        // 0..63
    const int n0     = blockIdx.y * 16;              // 0..1008
    const int m      = lane & 15;
    const int h      = lane >> 4;                    // half-wave: 0 or 1
    const int kbeg   = ksplit * KCHUNK;

    const v4f* ap = (const v4f*)(xb + (size_t)m * FC_IN + kbeg) + h;
    const v4f* wp = (const v4f*)(W1 + (size_t)(n0 + m) * FC_IN + kbeg) + h;

    v8f c = {};
    // 240 iterations, each: 4x global_load_b128 + 4x v_wmma (K-span 16)
    for (int i = 0; i < KCHUNK / 16; ++i) {
        v4f a0 = ap[4 * i],     b0 = wp[4 * i];
        v4f a1 = ap[4 * i + 2], b1 = wp[4 * i + 2];
        v2f ax, bx;
        ax.x = a0.x; ax.y = a0.y; bx.x = b0.x; bx.y = b0.y;
        c = WMMA_F32(ax, bx, c);
        ax.x = a0.z; ax.y = a0.w; bx.x = b0.z; bx.y = b0.w;
        c = WMMA_F32(ax, bx, c);
        ax.x = a1.x; ax.y = a1.y; bx.x = b1.x; bx.y = b1.y;
        c = WMMA_F32(ax, bx, c);
        ax.x = a1.z; ax.y = a1.w; bx.x = b1.z; bx.y = b1.w;
        c = WMMA_F32(ax, bx, c);
    }
    // C layout: VGPR r, lanes 0..15 -> (M=r, N=n0+lane). Only rows 0..3 are real.
    if (lane < 16) {
#pragma unroll
        for (int r = 0; r < BATCH; ++r)
            atomicAdd(&y1[r * 1024 + n0 + lane], c[r]);
    }
}

// relu + zero-pad y1 (4x1024) -> (16x1024)
__global__ void relu_pad_kernel(const float* __restrict__ y1, float* __restrict__ y1r)
{
    int idx = blockIdx.x * blockDim.x + threadIdx.x;
    if (idx >= MPAD * 1024) return;
    int row = idx >> 10;
    y1r[idx] = (row < BATCH) ? fmaxf(y1[idx], 0.f) : 0.f;
}

// FC2: (16x1024) * W2^T (128x1024) -> y2pre (4x128). One block, 8 waves, one n-tile each.
__global__ void fc2_wmma_kernel(const float* __restrict__ y1r,
                                const float* __restrict__ W2,
                                const float* __restrict__ b2,
                                float* __restrict__ y2)
{
    const int lane = threadIdx.x & 31;
    const int wave = threadIdx.x >> 5;
    const int n0   = wave * 16;
    const int m    = lane & 15;
    const int h    = lane >> 4;

    const v4f* ap = (const v4f*)(y1r + m * 1024) + h;
    const v4f* wp = (const v4f*)(W2 + (n0 + m) * 1024) + h;

    v8f c = {};
    for (int i = 0; i < 1024 / 16; ++i) {
        v4f a0 = ap[4 * i],     b0 = wp[4 * i];
        v4f a1 = ap[4 * i + 2], b1 = wp[4 * i + 2];
        v2f ax, bx;
        ax.x = a0.x; ax.y = a0.y; bx.x = b0.x; bx.y = b0.y;
        c = WMMA_F32(ax, bx, c);
        ax.x = a0.z; ax.y = a0.w; bx.x = b0.z; bx.y = b0.w;
        c = WMMA_F32(ax, bx, c);
        ax.x = a1.x; ax.y = a1.y; bx.x = b1.x; bx.y = b1.y;
        c = WMMA_F32(ax, bx, c);
        ax.x = a1.z; ax.y = a1.w; bx.x = b1.z; bx.y = b1.w;
        c = WMMA_F32(ax, bx, c);
    }
    if (lane < 16) {
#pragma unroll
        for (int r = 0; r < BATCH; ++r)
            y2[r * 128 + n0 + lane] = c[r] + b2[n0 + lane];
    }
}

// FC3: relu(y2pre) (4x128) * W3^T (10x128) + b3 -> out (4x10). Tiny; plain VALU.
__global__ void fc3_kernel(const float* __restrict__ y2,
                           const float* __restrict__ W3,
                           const float* __restrict__ b3,
                           float* __restrict__ out)
{
    int t = threadIdx.x;
    if (t >= BATCH * 10) return;
    int r = t / 10, n = t % 10;
    float acc = b3[n];
    const float* yr = y2 + r * 128;
    const float* wn = W3 + n * 128;
#pragma unroll 4
    for (int k = 0; k < 128; ++k)
        acc = fmaf(fmaxf(yr[k], 0.f), wn[k], acc);
    out[r * 10 + n] = acc;
}

// ---------------- launcher ----------------
extern "C" void kernel_launch(void* const* d_in, const int* in_sizes, int n_in,
                              void* d_out, int out_size, void* d_ws, size_t ws_size,
                              hipStream_t stream)
{
    const float* vin = (const float*)d_in[0];
    const float* syn = (const float*)d_in[1];
    const float* W1  = (const float*)d_in[2];
    const float* b1  = (const float*)d_in[3];
    const float* W2  = (const float*)d_in[4];
    const float* b2  = (const float*)d_in[5];
    const float* W3  = (const float*)d_in[6];
    const float* b3  = (const float*)d_in[7];
    float* out = (float*)d_out;

    // workspace layout (floats)
    float* ws   = (float*)d_ws;
    float* s0   = ws;                          // SVOL
    float* s1   = s0 + SVOL;                   // SVOL
    float* xb   = s1 + SVOL;                   // MPAD * FC_IN
    float* y1   = xb + (size_t)MPAD * FC_IN;   // 4 * 1024
    float* y1r  = y1 + BATCH * 1024;           // 16 * 1024
    float* y2   = y1r + MPAD * 1024;           // 4 * 128

    const int B = 256;

    // phase 1: recurrent field (5 steps, ping-pong)
    init_s_kernel<<<(SVOL + B - 1) / B, B, 0, stream>>>(vin, s0);
    float* sin_  = s0;
    float* sout_ = s1;
    for (int it = 0; it < 5; ++it) {
        step_kernel<<<(SVOL + B - 1) / B, B, 0, stream>>>(sin_, syn, sout_);
        float* tmp = sin_; sin_ = sout_; sout_ = tmp;
    }
    // final field is in sin_ after the swap

    // activation matrix for FC1
    make_x_kernel<<<(MPAD * FC_IN + B - 1) / B, B, 0, stream>>>(sin_, xb);

    // FC1 (WMMA, K-split, atomic reduce)
    fc1_init_kernel<<<(BATCH * 1024 + B - 1) / B, B, 0, stream>>>(b1, y1);
    fc1_wmma_kernel<<<dim3(8, 64), B, 0, stream>>>(xb, W1, y1);

    // FC2 (WMMA)
    relu_pad_kernel<<<(MPAD * 1024 + B - 1) / B, B, 0, stream>>>(y1, y1r);
    fc2_wmma_kernel<<<1, B, 0, stream>>>(y1r, W2, b2, y2);

    // FC3
    fc3_kernel<<<1, 64, 0, stream>>>(y2, W3, b3, out);
}